// EDDA_54803782697496
// MI455X (gfx1250) — compile-verified
//
#include <hip/hip_runtime.h>

// GRec cross-domain LightGCN forward for MI455X (gfx1250, wave32).
//
// Memory-bound scatter/gather pipeline; feature tables (<=64MB) are L2-resident
// on MI455X (192MB L2), so atomics/gathers hit L2 while edge lists stream from
// HBM (with global_prefetch). Final 8192 rowwise 64-dim dots are done with
// V_WMMA_F32_16X16X4_F32 (16 pairs/wave, diagonal extraction), full f32.

#define DIM 32
#define ALPHA 0.1f

typedef float v2f __attribute__((ext_vector_type(2)));
typedef float v8f __attribute__((ext_vector_type(8)));

static __device__ __forceinline__ const float*
rowptr(const float* fA, const float* fB, int nA, int node) {
  return (node < nA) ? (fA + (size_t)node * DIM)
                     : (fB + (size_t)(node - nA) * DIM);
}

// ---- zero n floats (n % 4 == 0) ------------------------------------------
__global__ __launch_bounds__(256) void gr_zero(float* __restrict__ p, long n) {
  long i = ((long)blockIdx.x * blockDim.x + threadIdx.x) * 4;
  if (i < n) *(float4*)(p + i) = make_float4(0.f, 0.f, 0.f, 0.f);
}

// ---- degree via atomic ones into dst -------------------------------------
__global__ __launch_bounds__(256) void gr_degree(const int* __restrict__ dst,
                                                 int nE, float* __restrict__ deg) {
  int e = blockIdx.x * blockDim.x + threadIdx.x;
  if (e < nE) atomicAdd(deg + dst[e], 1.0f);
}

// ---- deg -> rsqrt(max(deg,1)) in place -----------------------------------
__global__ __launch_bounds__(256) void gr_isr(float* __restrict__ deg, int n) {
  int i = blockIdx.x * blockDim.x + threadIdx.x;
  if (i < n) deg[i] = rsqrtf(fmaxf(deg[i], 1.0f));
}

// ---- edge scatter: agg[dst] += h[src] * isr[src]*isr[dst] ----------------
// 8 lanes per edge, float4 per lane -> one coalesced 128B row per edge.
__global__ __launch_bounds__(256) void gr_scatter(const int* __restrict__ src,
                                                  const int* __restrict__ dst,
                                                  int nE,
                                                  const float* __restrict__ isr,
                                                  const float* __restrict__ fA,
                                                  const float* __restrict__ fB,
                                                  int nA,
                                                  float* __restrict__ agg) {
  long t = (long)blockIdx.x * blockDim.x + threadIdx.x;
  long e = t >> 3;
  if (e >= nE) return;
  int c = (int)(t & 7) * 4;
  if (e + 4096 < nE) {                 // stream-prefetch edge arrays (global_prefetch_b8)
    __builtin_prefetch(src + e + 4096, 0, 1);
    __builtin_prefetch(dst + e + 4096, 0, 1);
  }
  int s = src[e], d = dst[e];
  float w = isr[s] * isr[d];
  const float* row = rowptr(fA, fB, nA, s);
  float4 v = *(const float4*)(row + c);
  float* out = agg + (size_t)d * DIM + c;
  atomicAdd(out + 0, v.x * w);
  atomicAdd(out + 1, v.y * w);
  atomicAdd(out + 2, v.z * w);
  atomicAdd(out + 3, v.w * w);
}

// ---- h = (1-alpha)*agg + alpha*x ; agg = 0 (ready for next layer) --------
__global__ __launch_bounds__(256) void gr_apply(float* __restrict__ h,
                                                float* __restrict__ agg,
                                                const float* __restrict__ xA,
                                                const float* __restrict__ xB,
                                                int nA, long n4) {
  long i = (long)blockIdx.x * blockDim.x + threadIdx.x;
  if (i >= n4) return;
  int node = (int)(i >> 3);            // DIM/4 == 8 float4 per row
  int c = (int)(i & 7) * 4;
  const float* xr = rowptr(xA, xB, nA, node);
  float4 x = *(const float4*)(xr + c);
  float4 a = ((const float4*)agg)[i];
  float4 r;
  r.x = (1.f - ALPHA) * a.x + ALPHA * x.x;
  r.y = (1.f - ALPHA) * a.y + ALPHA * x.y;
  r.z = (1.f - ALPHA) * a.z + ALPHA * x.z;
  r.w = (1.f - ALPHA) * a.w + ALPHA * x.w;
  ((float4*)h)[i] = r;
  ((float4*)agg)[i] = make_float4(0.f, 0.f, 0.f, 0.f);
}

// ---- light[r] += h[gathered node]/3 (mean over 3 domains folded in) ------
__global__ __launch_bounds__(256) void gr_accum(float* __restrict__ light,
                                                const float* __restrict__ h,
                                                const int* __restrict__ uids,
                                                const int* __restrict__ iids,
                                                int nDU, int nDom, int nTU) {
  long t = (long)blockIdx.x * blockDim.x + threadIdx.x;
  long r = t >> 3;
  if (r >= nDom) return;
  int c = (int)(t & 7) * 4;
  long node = (r < nDU) ? (long)uids[r] : (long)nTU + (long)iids[r - nDU];
  const float4 v = *(const float4*)(h + node * DIM + c);
  float4* dp = (float4*)(light + r * DIM + c);
  float4 acc = *dp;
  const float third = (1.0f / 3.0f);
  acc.x += v.x * third; acc.y += v.y * third;
  acc.z += v.z * third; acc.w += v.w * third;
  *dp = acc;
}

// ---- final: 16 pair-dots per wave via V_WMMA_F32_16X16X4_F32 -------------
// A = 16 user rows (16x64 f32), B = 16 item cols (64x16), D = A*B; gamma = diag.
// A frag layout (ISA 7.12.2): lanes 0-15 -> M=lane, vgpr{0,1}=K{0,1};
// lanes 16-31 -> vgpr{0,1}=K{2,3}. B mirrors with lane = column N.
// D(p,p): p<8 -> vgpr p @ lane p ; p>=8 -> vgpr p-8 @ lane p+16.
__global__ __launch_bounds__(256) void gr_gamma(const float* __restrict__ light,
                                                const float* __restrict__ hintra,
                                                const long long* __restrict__ users,
                                                const long long* __restrict__ items,
                                                float* __restrict__ gamma,
                                                int npairs, int nDU) {
  int wave = (int)blockIdx.x * (blockDim.x >> 5) + ((int)threadIdx.x >> 5);
  int lane = (int)threadIdx.x & 31;
  int pbase = wave * 16;
  if (pbase >= npairs) return;         // wave-uniform exit: EXEC stays all-ones
  int half = lane >> 4;
  int l = lane & 15;
  int p = pbase + l; if (p >= npairs) p = npairs - 1;
  long urow = (long)users[p];                    // user rows: [0, nDU)
  long irow = (long)nDU + (long)items[p];        // item rows: [nDU, nDom)
  const float* uL = light  + urow * DIM;  // inter half (dims 0..31)
  const float* uH = hintra + urow * DIM;  // intra half (dims 32..63)
  const float* iL = light  + irow * DIM;
  const float* iH = hintra + irow * DIM;
  v8f c = {0.f, 0.f, 0.f, 0.f, 0.f, 0.f, 0.f, 0.f};
  int o = half * 2;                     // K sub-offset within 4-wide slab
#pragma unroll
  for (int kk = 0; kk < 2 * DIM; kk += 4) {
    const float* ub = (kk < DIM) ? (uL + kk) : (uH + (kk - DIM));
    const float* ib = (kk < DIM) ? (iL + kk) : (iH + (kk - DIM));
    v2f a; a.x = ub[o]; a.y = ub[o + 1];
    v2f b; b.x = ib[o]; b.y = ib[o + 1];
    c = __builtin_amdgcn_wmma_f32_16x16x4_f32(
        /*neg_a=*/false, a, /*neg_b=*/false, b,
        /*c_mod=*/(short)0, c, /*reuse_a=*/false, /*reuse_b=*/false);
  }
  if (lane < 8) {
    if (pbase + lane < npairs) gamma[pbase + lane] = c[lane];
  } else if (lane >= 24) {
    int pp = pbase + lane - 16;
    if (pp < npairs) gamma[pp] = c[lane - 24];
  }
}

extern "C" void kernel_launch(void* const* d_in, const int* in_sizes, int n_in,
                              void* d_out, int out_size, void* d_ws, size_t ws_size,
                              hipStream_t stream) {
  const int*        dom_edges   = (const int*)d_in[0];
  const int*        sep_edges   = (const int*)d_in[1];
  const float*      aggr_user_w = (const float*)d_in[2];
  const float*      aggr_item_w = (const float*)d_in[3];
  const float*      dom_user_w  = (const float*)d_in[4];
  const float*      dom_item_w  = (const float*)d_in[5];
  const int*        uids        = (const int*)d_in[6];
  const int*        iids        = (const int*)d_in[7];
  const long long*  users       = (const long long*)d_in[8];
  const long long*  items       = (const long long*)d_in[9];
  float*            gamma       = (float*)d_out;

  const int nE_dom = in_sizes[0] / 2;
  const int nE_sep = in_sizes[1] / 6;
  const int nTU = in_sizes[2] / DIM, nTI = in_sizes[3] / DIM;
  const int nDU = in_sizes[4] / DIM, nDI = in_sizes[5] / DIM;
  const int nAll = nTU + nTI, nDom = nDU + nDI;
  const int npairs = in_sizes[8];

  // workspace layout (floats): agg[nAll*32] | h[nAll*32] | light[nDom*32] | deg[nAll]
  float* agg   = (float*)d_ws;
  float* h     = agg   + (size_t)nAll * DIM;
  float* light = h     + (size_t)nAll * DIM;
  float* deg   = light + (size_t)nDom * DIM;

  const int B = 256;
  auto blocks = [](long n) { return (int)((n + 255) / 256); };

  auto conv = [&](const int* src, const int* dst, int nE,
                  const float* xA, const float* xB, int nA, int nN) {
    long nElem = (long)nN * DIM;
    gr_zero   <<<blocks(nN / 4),      B, 0, stream>>>(deg, nN);
    gr_degree <<<blocks(nE),          B, 0, stream>>>(dst, nE, deg);
    gr_isr    <<<blocks(nN),          B, 0, stream>>>(deg, nN);
    gr_zero   <<<blocks(nElem / 4),   B, 0, stream>>>(agg, nElem);
    gr_scatter<<<blocks((long)nE * 8),B, 0, stream>>>(src, dst, nE, deg, xA, xB, nA, agg);
    gr_apply  <<<blocks(nElem / 4),   B, 0, stream>>>(h, agg, xA, xB, nA, nElem / 4);
    gr_scatter<<<blocks((long)nE * 8),B, 0, stream>>>(src, dst, nE, deg, h, h, nN, agg);
    gr_apply  <<<blocks(nElem / 4),   B, 0, stream>>>(h, agg, xA, xB, nA, nElem / 4);
  };

  // inter-domain: 3 convs on the 500K-node graph, mean folded into gr_accum
  gr_zero<<<blocks((long)nDom * DIM / 4), B, 0, stream>>>(light, (long)nDom * DIM);
  for (int d = 0; d < 3; ++d) {
    const int* src = sep_edges + (size_t)d * 2 * nE_sep;
    const int* dst = src + nE_sep;
    conv(src, dst, nE_sep, aggr_user_w, aggr_item_w, nTU, nAll);
    gr_accum<<<blocks((long)nDom * 8), B, 0, stream>>>(light, h, uids, iids, nDU, nDom, nTU);
  }

  // intra-domain conv LAST: its h stays live in `h` for the WMMA dot kernel
  conv(dom_edges, dom_edges + nE_dom, nE_dom, dom_user_w, dom_item_w, nDU, nDom);

  // final rowwise 64-dim dots via WMMA (16 pairs per wave)
  int nwaves = (npairs + 15) / 16;
  gr_gamma<<<(nwaves * 32 + B - 1) / B, B, 0, stream>>>(light, h, users, items,
                                                        gamma, npairs, nDU);
}